// Str2Str_44547400794149
// MI455X (gfx1250) — compile-verified
//
#include <hip/hip_runtime.h>
#include <hip/hip_bf16.h>

typedef __attribute__((ext_vector_type(2))) float v2f;
typedef __attribute__((ext_vector_type(8))) float v8f;

__device__ __forceinline__ v8f v8f_zero() {
    v8f z;
#pragma unroll
    for (int i = 0; i < 8; ++i) z[i] = 0.0f;
    return z;
}

// D = A(16x4 f32) * B(4x16 f32) + C(16x16 f32), wave32 WMMA
__device__ __forceinline__ v8f wmma_f32_4(v2f a, v2f b, v8f c) {
    return __builtin_amdgcn_wmma_f32_16x16x4_f32(false, a, false, b, (short)0, c, false, false);
}

// Async copy of 16 bytes: global -> LDS, tracked on ASYNCcnt (per-lane transfer).
__device__ __forceinline__ void async_g2l_b128(unsigned lds_addr, const void* gptr) {
    unsigned long long ga = (unsigned long long)(size_t)gptr;
    asm volatile("global_load_async_to_lds_b128 %0, %1, off" ::"v"(lds_addr), "v"(ga)
                 : "memory");
}
__device__ __forceinline__ void wait_asynccnt0() {
    asm volatile("s_wait_asynccnt 0" ::: "memory");
}
__device__ __forceinline__ unsigned lds_addr_of(const void* p) {
    return (unsigned)(size_t)p;  // generic->LDS: low 32 bits are the LDS offset
}

// Block-cooperative GEMM on resident (LDS or global) operands.
// Block = 128 threads = 4 waves. Wave w owns C rows [w*16, w*16+16).
// A: 64 x (4*ksteps) row-major, ldA.  B: (4*ksteps) x (NP16*16) row-major, ldB.
template <int NP16>
__device__ __forceinline__ void lds_gemm(const float* __restrict__ Ald, int ldA,
                                         const float* __restrict__ Bld, int ldB,
                                         int ksteps, v8f acc[NP16]) {
    const int lane = threadIdx.x & 31;
    const int wave = threadIdx.x >> 5;
    const int l = lane & 15;
    const int h = lane >> 4;
    const int arow = wave * 16 + l;
    for (int kk = 0; kk < ksteps; ++kk) {
        const int kb = kk * 4 + 2 * h;
        v2f a;
        a.x = Ald[arow * ldA + kb];
        a.y = Ald[arow * ldA + kb + 1];
#pragma unroll
        for (int j = 0; j < NP16; ++j) {
            v2f b;
            b.x = Bld[kb * ldB + j * 16 + l];
            b.y = Bld[(kb + 1) * ldB + j * 16 + l];
            acc[j] = wmma_f32_4(a, b, acc[j]);
        }
    }
}

// ---------------------------------------------------------------------------
// Generic row LayerNorm (no affine).  grid = rows, block = 256.
// ---------------------------------------------------------------------------
__global__ void ln_rows_kernel(const float* __restrict__ in, float* __restrict__ out,
                               int width, int in_stride, int out_stride) {
    __shared__ float red[256];
    const int row = blockIdx.x;
    const int tid = threadIdx.x;
    const float* src = in + (size_t)row * in_stride;
    float* dst = out + (size_t)row * out_stride;

    float s = 0.0f;
    for (int c = tid; c < width; c += 256) s += src[c];
    red[tid] = s;
    __syncthreads();
    for (int off = 128; off > 0; off >>= 1) {
        if (tid < off) red[tid] += red[tid + off];
        __syncthreads();
    }
    const float mean = red[0] / (float)width;
    __syncthreads();
    float s2 = 0.0f;
    for (int c = tid; c < width; c += 256) {
        float d = src[c] - mean;
        s2 += d * d;
    }
    red[tid] = s2;
    __syncthreads();
    for (int off = 128; off > 0; off >>= 1) {
        if (tid < off) red[tid] += red[tid + off];
        __syncthreads();
    }
    const float rstd = rsqrtf(red[0] / (float)width + 1e-5f);
    __syncthreads();
    for (int c = tid; c < width; c += 256) dst[c] = (src[c] - mean) * rstd;
}

// ---------------------------------------------------------------------------
// Top-k (k=64) nearest neighbors per residue, jax.lax.top_k tie-breaking
// (smallest Dmod, ties -> lowest index).  grid = L, block = 256.
// ---------------------------------------------------------------------------
__global__ void topk_kernel(const float* __restrict__ xyz, const int* __restrict__ idx,
                            int* __restrict__ nbr, float* __restrict__ Dsel) {
    __shared__ float Drow[1024];
    __shared__ float Key[1024];
    __shared__ float rv[256];
    __shared__ int ri[256];
    const int i = blockIdx.x;
    const int tid = threadIdx.x;
    const float cix = xyz[(size_t)i * 9 + 3 + 0];
    const float ciy = xyz[(size_t)i * 9 + 3 + 1];
    const float ciz = xyz[(size_t)i * 9 + 3 + 2];
    const int ii = idx[i];

    for (int j = tid; j < 1024; j += 256) {
        float dx = xyz[(size_t)j * 9 + 3 + 0] - cix;
        float dy = xyz[(size_t)j * 9 + 3 + 1] - ciy;
        float dz = xyz[(size_t)j * 9 + 3 + 2] - ciz;
        float sq = dx * dx + dy * dy + dz * dz;
        float d = sqrtf(fmaxf(sq, 1e-12f));
        Drow[j] = d;
        int sep = idx[j] - ii;
        float asep = fabsf((float)sep);
        float eye = (i == j) ? 999.9f : 0.0f;
        Key[j] = d + eye + (asep + eye) * 1e-6f;
    }
    __syncthreads();

    for (int it = 0; it < 64; ++it) {
        float bv = 1e30f;
        int bi = 1 << 20;
        for (int j = tid; j < 1024; j += 256) {
            float v = Key[j];
            if (v < bv || (v == bv && j < bi)) { bv = v; bi = j; }
        }
        rv[tid] = bv;
        ri[tid] = bi;
        __syncthreads();
        for (int off = 128; off > 0; off >>= 1) {
            if (tid < off) {
                float ov = rv[tid + off];
                int oi = ri[tid + off];
                if (ov < rv[tid] || (ov == rv[tid] && oi < ri[tid])) {
                    rv[tid] = ov;
                    ri[tid] = oi;
                }
            }
            __syncthreads();
        }
        if (tid == 0) {
            int jm = ri[0];
            nbr[(size_t)i * 64 + it] = jm;
            Dsel[(size_t)i * 64 + it] = Drow[jm];
            Key[jm] = 1e30f;
        }
        __syncthreads();
    }
}

// ---------------------------------------------------------------------------
// Edge embedding: only at the 64 gathered neighbors of each residue.
// Pair rows are gathered with GLOBAL_LOAD_ASYNC_TO_LDS_B128 (one 512B row per
// wave-instruction, no VGPR round trip), overlapped with w_e2 staging.
// grid = L, block = 128 (4 waves).
// ---------------------------------------------------------------------------
__global__ void edge_kernel(const float* __restrict__ pair, const float* __restrict__ w_e1,
                            const float* __restrict__ b_e1, const float* __restrict__ w_e2,
                            const float* __restrict__ b_e2, const int* __restrict__ nbr,
                            const float* __restrict__ Dsel, const int* __restrict__ idx,
                            float* __restrict__ edge_out, int L) {
    __shared__ float P[64 * 128];   // gathered pair rows (async dst); later reused as E (64x32)
    __shared__ float Z[64 * 72];    // ln(Y) | rbf | seqsep | zero-pad (K padded 69->72)
    __shared__ float B2[72 * 32];   // zero-padded w_e2
    __shared__ int nb[64];
    const int i = blockIdx.x;
    const int tid = threadIdx.x;
    const int lane = tid & 31;
    const int wave = tid >> 5;

    if (tid < 64) nb[tid] = nbr[(size_t)i * 64 + tid];
    __syncthreads();

    // async gather: wave w moves rows w, w+4, ... (each lane 16B)
    for (int r = wave; r < 64; r += 4) {
        const float* src = pair + ((size_t)i * L + (size_t)nb[r]) * 128 + lane * 4;
        async_g2l_b128(lds_addr_of(&P[r * 128 + lane * 4]), src);
    }
    // overlap: stage zero-padded w_e2 while the gather is in flight
    for (int e = tid; e < 72 * 32; e += 128) {
        int k = e >> 5, n = e & 31;
        B2[e] = (k < 69) ? w_e2[k * 32 + n] : 0.0f;
    }
    wait_asynccnt0();
    __syncthreads();

    // LN(128) in place on the gathered rows
    for (int r = wave; r < 64; r += 4) {
        float x[4];
        float s = 0.0f;
#pragma unroll
        for (int q = 0; q < 4; ++q) { x[q] = P[r * 128 + lane + 32 * q]; s += x[q]; }
        for (int m = 16; m > 0; m >>= 1) s += __shfl_xor(s, m, 32);
        float mean = s * (1.0f / 128.0f);
        float s2 = 0.0f;
#pragma unroll
        for (int q = 0; q < 4; ++q) { float d = x[q] - mean; s2 += d * d; }
        for (int m = 16; m > 0; m >>= 1) s2 += __shfl_xor(s2, m, 32);
        float rstd = rsqrtf(s2 * (1.0f / 128.0f) + 1e-5f);
#pragma unroll
        for (int q = 0; q < 4; ++q) P[r * 128 + lane + 32 * q] = (x[q] - mean) * rstd;
    }
    __syncthreads();

    // GEMM1: P(64x128) @ w_e1(128x32)   (w_e1 streamed from global / L2)
    v8f acc[2];
    acc[0] = v8f_zero();
    acc[1] = v8f_zero();
    lds_gemm<2>(P, 128, w_e1, 32, 32, acc);
    {
        const int l = lane & 15, h = lane >> 4;
#pragma unroll
        for (int j = 0; j < 2; ++j) {
            int col = j * 16 + l;
#pragma unroll
            for (int v = 0; v < 8; ++v) {
                int r = wave * 16 + v + 8 * h;
                Z[r * 72 + col] = acc[j][v] + b_e1[col];
            }
        }
    }
    __syncthreads();

    // LN(32) of Y in place (cols 0..31), plus fill rbf/seqsep cols 32..71.
    for (int r = wave; r < 64; r += 4) {
        float x = Z[r * 72 + lane];
        float s = x;
        for (int m = 16; m > 0; m >>= 1) s += __shfl_xor(s, m, 32);
        float mean = s * (1.0f / 32.0f);
        float d = x - mean;
        float s2 = d * d;
        for (int m = 16; m > 0; m >>= 1) s2 += __shfl_xor(s2, m, 32);
        float rstd = rsqrtf(s2 * (1.0f / 32.0f) + 1e-5f);
        Z[r * 72 + lane] = d * rstd;
    }
    const int ii = idx[i];
    for (int e = tid; e < 64 * 40; e += 128) {
        int r = e / 40, c = e % 40;   // c -> column 32+c
        float val = 0.0f;
        if (c < 36) {
            float d = Dsel[(size_t)i * 64 + r];
            float mu = 20.0f * (float)c / 35.0f;
            float t = (d - mu) * (36.0f / 20.0f);
            val = __expf(-t * t);
        } else if (c == 36) {
            int sep = idx[nb[r]] - ii;
            int as = sep < 0 ? -sep : sep;
            float mag = (as > 1) ? 0.0f : (float)as;
            float sgn = (sep > 0) ? 1.0f : ((sep < 0) ? -1.0f : 0.0f);
            val = sgn * mag;
        }
        Z[r * 72 + 32 + c] = val;
    }
    __syncthreads();

    // GEMM2: Z(64x72) @ B2(72x32)
    acc[0] = v8f_zero();
    acc[1] = v8f_zero();
    lds_gemm<2>(Z, 72, B2, 32, 18, acc);
    float* E = P;  // reuse
    {
        const int l = lane & 15, h = lane >> 4;
#pragma unroll
        for (int j = 0; j < 2; ++j) {
            int col = j * 16 + l;
#pragma unroll
            for (int v = 0; v < 8; ++v) {
                int r = wave * 16 + v + 8 * h;
                E[r * 32 + col] = acc[j][v] + b_e2[col];
            }
        }
    }
    __syncthreads();

    // LN(32) + write out
    for (int r = wave; r < 64; r += 4) {
        float x = E[r * 32 + lane];
        float s = x;
        for (int m = 16; m > 0; m >>= 1) s += __shfl_xor(s, m, 32);
        float mean = s * (1.0f / 32.0f);
        float d = x - mean;
        float s2 = d * d;
        for (int m = 16; m > 0; m >>= 1) s2 += __shfl_xor(s2, m, 32);
        float rstd = rsqrtf(s2 * (1.0f / 32.0f) + 1e-5f);
        edge_out[((size_t)i * 64 + r) * 32 + lane] = d * rstd;
    }
}

// ---------------------------------------------------------------------------
// Neighbor attention + frame update.  h_j and edge rows fetched with async
// global->LDS B128 transfers overlapped with Wk/We staging and q-vector.
// grid = L, block = 128 (4 waves).
// ---------------------------------------------------------------------------
__global__ void attn_kernel(const float* __restrict__ node, const float* __restrict__ edge,
                            const int* __restrict__ nbr, const float* __restrict__ xyz,
                            const float* __restrict__ Wq, const float* __restrict__ b_m,
                            const float* __restrict__ Wk, const float* __restrict__ We,
                            const float* __restrict__ Wa, const float* __restrict__ Wl1,
                            const float* __restrict__ R_in, const float* __restrict__ T_in,
                            const unsigned char* __restrict__ motif_mask,
                            float* __restrict__ msum, float* __restrict__ Ri_out,
                            float* __restrict__ Ti_out) {
    __shared__ float HJ[64 * 32];
    __shared__ float EG[64 * 32];
    __shared__ float BK[32 * 32];
    __shared__ float BE[32 * 32];
    __shared__ float Mv[64 * 32];
    __shared__ float qb[32];
    __shared__ float contrib[64 * 6];
    __shared__ int nb[64];
    __shared__ float ni[32];
    const int i = blockIdx.x;
    const int tid = threadIdx.x;
    const int lane = tid & 31;
    const int wave = tid >> 5;

    if (tid < 64) nb[tid] = nbr[(size_t)i * 64 + tid];
    if (tid < 32) ni[tid] = node[(size_t)i * 32 + tid];
    __syncthreads();

    // async gathers: 16B chunks; HJ row r = node[nb[r]] (128B = 8 chunks/row),
    // EG = contiguous 8KB block edge[i*64*32 ...]
    for (int e = tid; e < 512; e += 128) {
        int r = e >> 3, sub = e & 7;
        const float* hsrc = node + (size_t)nb[r] * 32 + sub * 4;
        async_g2l_b128(lds_addr_of(&HJ[e * 4]), hsrc);
        const float* esrc = edge + (size_t)i * 64 * 32 + e * 4;
        async_g2l_b128(lds_addr_of(&EG[e * 4]), esrc);
    }
    // overlap: stage Wk/We and compute q-vector while transfers are in flight
    for (int e = tid; e < 32 * 32; e += 128) {
        BK[e] = Wk[e];
        BE[e] = We[e];
    }
    if (tid < 32) {
        float s = b_m[tid];
        for (int k = 0; k < 32; ++k) s += ni[k] * Wq[k * 32 + tid];
        qb[tid] = s;
    }
    wait_asynccnt0();
    __syncthreads();

    v8f acc[2];
    acc[0] = v8f_zero();
    acc[1] = v8f_zero();
    lds_gemm<2>(HJ, 32, BK, 32, 8, acc);
    lds_gemm<2>(EG, 32, BE, 32, 8, acc);
    {
        const int l = lane & 15, h = lane >> 4;
#pragma unroll
        for (int j = 0; j < 2; ++j) {
            int col = j * 16 + l;
#pragma unroll
            for (int v = 0; v < 8; ++v) {
                int r = wave * 16 + v + 8 * h;
                Mv[r * 32 + col] = fmaxf(acc[j][v] + qb[col], 0.0f);
            }
        }
    }
    __syncthreads();

    if (tid < 32) {
        float s = 0.0f;
        for (int r = 0; r < 64; ++r) s += Mv[r * 32 + tid];
        msum[(size_t)i * 32 + tid] = s * (1.0f / 64.0f);   // mean over k
    }
    const float cix = xyz[(size_t)i * 9 + 3 + 0];
    const float ciy = xyz[(size_t)i * 9 + 3 + 1];
    const float ciz = xyz[(size_t)i * 9 + 3 + 2];
    if (tid < 64) {
        int r = tid;
        float a0 = 0.0f, a1 = 0.0f;
        for (int c = 0; c < 32; ++c) {
            float mv = Mv[r * 32 + c];
            a0 += mv * Wa[c * 2 + 0];
            a1 += mv * Wa[c * 2 + 1];
        }
        int j = nb[r];
        float rx = xyz[(size_t)j * 9 + 3 + 0] - cix;
        float ry = xyz[(size_t)j * 9 + 3 + 1] - ciy;
        float rz = xyz[(size_t)j * 9 + 3 + 2] - ciz;
        contrib[r * 6 + 0] = a0 * rx;
        contrib[r * 6 + 1] = a0 * ry;
        contrib[r * 6 + 2] = a0 * rz;
        contrib[r * 6 + 3] = a1 * rx;
        contrib[r * 6 + 4] = a1 * ry;
        contrib[r * 6 + 5] = a1 * rz;
    }
    __syncthreads();

    if (tid == 0) {
        float off[2][3];
        for (int c = 0; c < 2; ++c)
            for (int v = 0; v < 3; ++v) {
                float s = 0.0f;
                for (int r = 0; r < 64; ++r) s += contrib[r * 6 + c * 3 + v];
                off[c][v] = s * (1.0f / 64.0f);
            }
        for (int c = 0; c < 2; ++c)
            for (int v = 0; v < 3; ++v) {
                float s = 0.0f;
                for (int at = 0; at < 3; ++at) {
                    float l1 = xyz[(size_t)i * 9 + at * 3 + v] - xyz[(size_t)i * 9 + 3 + v];
                    s += Wl1[c * 3 + at] * l1;
                }
                off[c][v] += s;
            }
        if (motif_mask[i]) {
            for (int c = 0; c < 2; ++c)
                for (int v = 0; v < 3; ++v) off[c][v] = 0.0f;
        }
        float dT[3] = {off[0][0] * 0.1f, off[0][1] * 0.1f, off[0][2] * 0.1f};
        float Rx = off[1][0] * 0.01f, Ry = off[1][1] * 0.01f, Rz = off[1][2] * 0.01f;
        float Qn = sqrtf(1.0f + Rx * Rx + Ry * Ry + Rz * Rz);
        float qA = 1.0f / Qn, qB = Rx / Qn, qC = Ry / Qn, qD = Rz / Qn;
        float dR[3][3];
        dR[0][0] = qA * qA + qB * qB - qC * qC - qD * qD;
        dR[0][1] = 2 * qB * qC - 2 * qA * qD;
        dR[0][2] = 2 * qB * qD + 2 * qA * qC;
        dR[1][0] = 2 * qB * qC + 2 * qA * qD;
        dR[1][1] = qA * qA - qB * qB + qC * qC - qD * qD;
        dR[1][2] = 2 * qC * qD - 2 * qA * qB;
        dR[2][0] = 2 * qB * qD - 2 * qA * qC;
        dR[2][1] = 2 * qC * qD + 2 * qA * qB;
        dR[2][2] = qA * qA - qB * qB - qC * qC + qD * qD;
        for (int r = 0; r < 3; ++r)
            for (int c = 0; c < 3; ++c) {
                float s = 0.0f;
                for (int k = 0; k < 3; ++k) s += dR[r][k] * R_in[(size_t)i * 9 + k * 3 + c];
                Ri_out[(size_t)i * 9 + r * 3 + c] = s;
            }
        for (int v = 0; v < 3; ++v) Ti_out[(size_t)i * 3 + v] = dT[v] + T_in[(size_t)i * 3 + v];
    }
}

// ---------------------------------------------------------------------------
// Generic global-memory GEMM, f32 WMMA.  C = epi(A@B + bias + Cadd).
// NP16 = N_padded/16.  Block = 128 threads (4 waves) over 64 rows.
// ---------------------------------------------------------------------------
template <int NP16>
__global__ void gemm_wmma_kernel(const float* __restrict__ A, int lda,
                                 const float* __restrict__ B, int ldb,  // ldb = N_real
                                 const float* __restrict__ bias, const float* __restrict__ Cadd,
                                 float* __restrict__ Cout, int ldc, int M, int Nreal, int K,
                                 int relu_a, int relu_out) {
    constexpr int NPAD = NP16 * 16;
    constexpr int CHUNK = 32;
    __shared__ float Ald[64 * CHUNK];
    __shared__ float Bld[CHUNK * NPAD];
    const int row0 = blockIdx.x * 64;
    const int tid = threadIdx.x;

    v8f acc[NP16];
#pragma unroll
    for (int j = 0; j < NP16; ++j) acc[j] = v8f_zero();

    for (int k0 = 0; k0 < K; k0 += CHUNK) {
        for (int e = tid; e < 64 * CHUNK; e += 128) {
            int r = e / CHUNK, kc = e % CHUNK;
            int gr = row0 + r, gk = k0 + kc;
            float v = 0.0f;
            if (gr < M && gk < K) v = A[(size_t)gr * lda + gk];
            if (relu_a) v = fmaxf(v, 0.0f);
            Ald[r * CHUNK + kc] = v;
        }
        for (int e = tid; e < CHUNK * NPAD; e += 128) {
            int kc = e / NPAD, n = e % NPAD;
            int gk = k0 + kc;
            float v = 0.0f;
            if (gk < K && n < Nreal) v = B[(size_t)gk * ldb + n];
            Bld[kc * NPAD + n] = v;
        }
        __syncthreads();
        lds_gemm<NP16>(Ald, CHUNK, Bld, NPAD, CHUNK / 4, acc);
        __syncthreads();
    }

    const int lane = tid & 31, wave = tid >> 5;
    const int l = lane & 15, h = lane >> 4;
#pragma unroll
    for (int j = 0; j < NP16; ++j) {
        int col = j * 16 + l;
#pragma unroll
        for (int v = 0; v < 8; ++v) {
            int row = row0 + wave * 16 + v + 8 * h;
            if (row < M && col < Nreal) {
                float val = acc[j][v];
                if (bias) val += bias[col];
                if (Cadd) val += Cadd[(size_t)row * ldc + col];
                if (relu_out) val = fmaxf(val, 0.0f);
                Cout[(size_t)row * ldc + col] = val;
            }
        }
    }
}

// ---------------------------------------------------------------------------
// Host launch
// ---------------------------------------------------------------------------
extern "C" void kernel_launch(void* const* d_in, const int* in_sizes, int n_in, void* d_out,
                              int out_size, void* d_ws, size_t ws_size, hipStream_t stream) {
    (void)in_sizes; (void)n_in; (void)out_size; (void)ws_size;
    const int L = 1024;

    const float* msa = (const float*)d_in[0];
    const float* pair = (const float*)d_in[1];
    const float* R_in = (const float*)d_in[2];
    const float* T_in = (const float*)d_in[3];
    const float* xyz = (const float*)d_in[4];
    const float* state = (const float*)d_in[5];
    const int* idx = (const int*)d_in[6];
    const unsigned char* mmask = (const unsigned char*)d_in[7];
    // d_in[8] = top_k (fixed at 64)
    // params dict, insertion order:
    const float* w_ex = (const float*)d_in[9];
    const float* b_ex = (const float*)d_in[10];
    const float* w_e1 = (const float*)d_in[11];
    const float* b_e1 = (const float*)d_in[12];
    const float* w_e2 = (const float*)d_in[13];
    const float* b_e2 = (const float*)d_in[14];
    const float* Wq = (const float*)d_in[15];
    const float* b_m = (const float*)d_in[16];
    const float* Wk = (const float*)d_in[17];
    const float* We = (const float*)d_in[18];
    const float* W0 = (const float*)d_in[19];
    const float* b_0 = (const float*)d_in[20];
    const float* Ws = (const float*)d_in[21];
    const float* Wa = (const float*)d_in[22];
    const float* Wl1 = (const float*)d_in[23];
    const float* w_s0 = (const float*)d_in[24];
    const float* b_s0 = (const float*)d_in[25];
    const float* w_si = (const float*)d_in[26];
    const float* b_si = (const float*)d_in[27];
    const float* w1 = (const float*)d_in[28];
    const float* b1 = (const float*)d_in[29];
    const float* w2 = (const float*)d_in[30];
    const float* b2 = (const float*)d_in[31];
    const float* w3 = (const float*)d_in[32];
    const float* b3 = (const float*)d_in[33];
    const float* w4 = (const float*)d_in[34];
    const float* b4 = (const float*)d_in[35];
    const float* w_out = (const float*)d_in[36];
    const float* b_out = (const float*)d_in[37];

    // workspace layout (floats)
    float* f = (float*)d_ws;
    float* lnseq = f;                       // 1024*256
    float* lnstate = lnseq + 262144;        // 1024*16
    float* nodepre = lnstate + 16384;       // 1024*32
    float* node = nodepre + 32768;          // 1024*32
    float* Dselb = node + 32768;            // 1024*64
    float* edge = Dselb + 65536;            // 1024*64*32
    float* msum = edge + 2097152;           // 1024*32
    float* state_out = msum + 32768;        // 1024*16
    float* lnso = state_out + 16384;        // 1024*16
    float* si = lnso + 16384;               // 1024*128
    float* tbuf = si + 131072;              // 1024*128
    int* nbr = (int*)(tbuf + 131072);       // 1024*64 ints

    float* out = (float*)d_out;
    float* Ri_out = out;            // 1024*9
    float* Ti_out = out + 9216;     // 1024*3
    float* so_out = out + 12288;    // 1024*16
    float* alpha = out + 28672;     // 1024*20

    // node features
    ln_rows_kernel<<<L, 256, 0, stream>>>(msa, lnseq, 256, 256, 256);
    ln_rows_kernel<<<L, 256, 0, stream>>>(state, lnstate, 16, 16, 16);
    gemm_wmma_kernel<2><<<16, 128, 0, stream>>>(lnseq, 256, w_ex, 32, b_ex, nullptr, nodepre, 32,
                                                L, 32, 256, 0, 0);
    gemm_wmma_kernel<2><<<16, 128, 0, stream>>>(lnstate, 16, w_ex + 256 * 32, 32, nullptr,
                                                nodepre, nodepre, 32, L, 32, 16, 0, 0);
    ln_rows_kernel<<<L, 256, 0, stream>>>(nodepre, node, 32, 32, 32);

    // neighbors first -> edge embedding only at gathered pairs (16x less pair traffic)
    topk_kernel<<<L, 256, 0, stream>>>(xyz, idx, nbr, Dselb);
    edge_kernel<<<L, 128, 0, stream>>>(pair, w_e1, b_e1, w_e2, b_e2, nbr, Dselb, idx, edge, L);

    // attention + frame update (writes Ri, Ti)
    attn_kernel<<<L, 128, 0, stream>>>(node, edge, nbr, xyz, Wq, b_m, Wk, We, Wa, Wl1, R_in,
                                       T_in, mmask, msum, Ri_out, Ti_out);

    // state_out = mean(m)@W0 + b_0 + node@Ws
    gemm_wmma_kernel<1><<<16, 128, 0, stream>>>(msum, 32, W0, 16, b_0, nullptr, state_out, 16, L,
                                                16, 32, 0, 0);
    gemm_wmma_kernel<1><<<16, 128, 0, stream>>>(node, 32, Ws, 16, nullptr, state_out, state_out,
                                                16, L, 16, 32, 0, 0);
    hipMemcpyAsync(so_out, state_out, (size_t)L * 16 * sizeof(float), hipMemcpyDeviceToDevice,
                   stream);

    // sc_pred
    ln_rows_kernel<<<L, 256, 0, stream>>>(state_out, lnso, 16, 16, 16);
    gemm_wmma_kernel<8><<<16, 128, 0, stream>>>(lnseq, 256, w_s0, 128, b_s0, nullptr, si, 128, L,
                                                128, 256, 0, 0);
    gemm_wmma_kernel<8><<<16, 128, 0, stream>>>(lnso, 16, w_si, 128, b_si, si, si, 128, L, 128,
                                                16, 0, 0);
    gemm_wmma_kernel<8><<<16, 128, 0, stream>>>(si, 128, w1, 128, b1, nullptr, tbuf, 128, L, 128,
                                                128, 1, 1);
    gemm_wmma_kernel<8><<<16, 128, 0, stream>>>(tbuf, 128, w2, 128, b2, si, si, 128, L, 128, 128,
                                                0, 0);
    gemm_wmma_kernel<8><<<16, 128, 0, stream>>>(si, 128, w3, 128, b3, nullptr, tbuf, 128, L, 128,
                                                128, 1, 1);
    gemm_wmma_kernel<8><<<16, 128, 0, stream>>>(tbuf, 128, w4, 128, b4, si, si, 128, L, 128, 128,
                                                0, 0);
    gemm_wmma_kernel<2><<<16, 128, 0, stream>>>(si, 128, w_out, 20, b_out, nullptr, alpha, 20, L,
                                                20, 128, 1, 0);
}